// CUF_6390911336554
// MI455X (gfx1250) — compile-verified
//
#include <hip/hip_runtime.h>
#include <hip/hip_bf16.h>

// ---------------------------------------------------------------------------
// Problem constants (from reference)
// ---------------------------------------------------------------------------
#define BATCH 4
#define CH    64
#define HIN   96
#define WIN   96
#define HT    192
#define WT    192
#define TILE  16          // output tile per workgroup (16x16 pixels)

typedef __attribute__((ext_vector_type(2))) float v2f;
typedef __attribute__((ext_vector_type(4))) float v4f;
typedef __attribute__((ext_vector_type(8))) float v8f;

// ---------------------------------------------------------------------------
// Kernel 0: transpose feat (B,64,96,96) -> featT (B,96,96,64)  (channels last)
// LDS-tiled so both global read and write are coalesced.
// grid = (3, 96, B), block = 256
// ---------------------------------------------------------------------------
__global__ __launch_bounds__(256) void k_transpose_feat(
    const float* __restrict__ feat, float* __restrict__ featT)
{
    __shared__ float tile[64][33];          // 64 ch x 32 x, padded
    const int tid = threadIdx.x;
    const int x0  = blockIdx.x * 32;
    const int y   = blockIdx.y;
    const int b   = blockIdx.z;

    // load: 2048 elements, x fastest across lanes -> coalesced reads
    #pragma unroll
    for (int j = 0; j < 8; ++j) {
        int id = tid + j * 256;
        int c  = id >> 5;
        int xx = id & 31;
        tile[c][xx] = feat[((size_t)(b * CH + c) * HIN + y) * WIN + x0 + xx];
    }
    __syncthreads();
    // store: c fastest across lanes -> coalesced writes; LDS stride 33 -> no conflicts
    #pragma unroll
    for (int j = 0; j < 8; ++j) {
        int id = tid + j * 256;
        int xx = id >> 6;
        int c  = id & 63;
        featT[((size_t)(b * HIN + y) * WIN + x0 + xx) * CH + c] = tile[c][xx];
    }
}

// ---------------------------------------------------------------------------
// Kernel 1: fused  sample * kernel -> fold -> (w1 matmul via WMMA) -> leaky
// One workgroup = one 16x16 tile of target pixels for one batch.
//   Phase A: 18x18 halo of target pixels; each computes its 4x4 feat patch
//            (float4 over channels), forms the 9 tap samples, multiplies by
//            the predicted kernel, scatters into LDS folded[64][256] (ds_add).
//   Phase B: h = leakyrelu(w1 x folded + b1) with v_wmma_f32_16x16x4_f32,
//            results staged in LDS, coalesced store to hws (B,192,192,64).
// grid = (12, 12, B), block = 256 (8 waves)
// ---------------------------------------------------------------------------
__global__ __launch_bounds__(256) void k_fused_main(
    const float* __restrict__ featT,   // (B,96,96,64)
    const float* __restrict__ kern,    // (B,576,36864)
    const float* __restrict__ coord,   // (B,192,192,2)
    const float* __restrict__ w1,      // (64,64) row-major [o][c]
    const float* __restrict__ b1,      // (64)
    float* __restrict__ hws)           // (B,192,192,64)
{
    __shared__ float s_fold[CH * 256];      // folded[c][pix]   64 KB
    __shared__ float s_h[256 * 68];         // h[pix][o] pad->68 68 KB

    const int tid = threadIdx.x;
    const int b   = blockIdx.z;
    const int th0 = blockIdx.y * TILE;
    const int tw0 = blockIdx.x * TILE;

    for (int i = tid; i < CH * 256; i += 256) s_fold[i] = 0.0f;
    __syncthreads();

    const float* kbase = kern  + (size_t)b * (576u * (HT * WT));
    const float* fbase = featT + (size_t)b * (HIN * WIN * CH);

    // ---------------- Phase A: scatter over 18x18 halo --------------------
    for (int hp = tid; hp < 18 * 18; hp += 256) {
        const int lh = hp / 18, lw = hp % 18;
        const int hh = th0 + lh - 1;          // target pixel h'
        const int ww = tw0 + lw - 1;          // target pixel w'
        if (hh < 0 || hh >= HT || ww < 0 || ww >= WT) continue;
        const int pos = hh * WT + ww;

        // grid = sample_coord[..., ::-1] : x = coord[...,1], y = coord[...,0]
        const float sy = coord[((size_t)(b * HT + hh) * WT + ww) * 2 + 0];
        const float sx = coord[((size_t)(b * HT + hh) * WT + ww) * 2 + 1];
        const float ix = ((sx + 1.0f) * (float)WIN - 1.0f) * 0.5f;
        const float iy = ((sy + 1.0f) * (float)HIN - 1.0f) * 0.5f;
        const float fx0 = floorf(ix), fy0 = floorf(iy);
        const float wx1 = ix - fx0,  wy1 = iy - fy0;
        const float wx0 = 1.0f - wx1, wy0 = 1.0f - wy1;
        const int x0 = (int)fx0, y0 = (int)fy0;     // in [-1, 95]
        const bool vx0 = (x0 >= 0) && (x0 < WIN);
        const bool vx1 = (x0 + 1 >= 0) && (x0 + 1 < WIN);
        const bool vy0 = (y0 >= 0) && (y0 < HIN);
        const bool vy1 = (y0 + 1 >= 0) && (y0 + 1 < HIN);
        float Wc[2][2];
        Wc[0][0] = wy0 * wx0 * (float)(vy0 && vx0);
        Wc[0][1] = wy0 * wx1 * (float)(vy0 && vx1);
        Wc[1][0] = wy1 * wx0 * (float)(vy1 && vx0);
        Wc[1][1] = wy1 * wx1 * (float)(vy1 && vx1);

        // destination (local folded index) per tap, -1 if outside tile
        int dest[9];
        #pragma unroll
        for (int ki = 0; ki < 3; ++ki)
            #pragma unroll
            for (int kj = 0; kj < 3; ++kj) {
                const int hd = lh + ki - 2, wd = lw + kj - 2;
                dest[ki * 3 + kj] =
                    (hd >= 0 && hd < TILE && wd >= 0 && wd < TILE)
                        ? hd * TILE + wd : -1;
            }

        const float* kp = kbase + pos;

        for (int c0 = 0; c0 < CH; c0 += 4) {
            // 4x4 spatial patch, 4 channels per float4 (channels-last layout)
            v4f P[16];
            #pragma unroll
            for (int dy = 0; dy < 4; ++dy) {
                const int fy = y0 - 1 + dy;
                const bool vy = (fy >= 0) && (fy < HIN);
                #pragma unroll
                for (int dx = 0; dx < 4; ++dx) {
                    const int fxp = x0 - 1 + dx;
                    v4f val = {0.0f, 0.0f, 0.0f, 0.0f};
                    if (vy && fxp >= 0 && fxp < WIN)
                        val = *(const v4f*)(fbase +
                              ((size_t)(fy * WIN + fxp)) * CH + c0);
                    P[dy * 4 + dx] = val;
                }
            }
            #pragma unroll
            for (int cc = 0; cc < 4; ++cc) {
                const float* kc = kp + (size_t)(c0 + cc) * 9 * (HT * WT);
                #pragma unroll
                for (int ki = 0; ki < 3; ++ki)
                    #pragma unroll
                    for (int kj = 0; kj < 3; ++kj) {
                        const int d = dest[ki * 3 + kj];
                        if (d >= 0) {
                            const float s =
                                Wc[0][0] * P[ki * 4 + kj][cc] +
                                Wc[0][1] * P[ki * 4 + kj + 1][cc] +
                                Wc[1][0] * P[(ki + 1) * 4 + kj][cc] +
                                Wc[1][1] * P[(ki + 1) * 4 + kj + 1][cc];
                            const float pr =
                                s * kc[(size_t)(ki * 3 + kj) * (HT * WT)];
                            atomicAdd(&s_fold[(c0 + cc) * 256 + d], pr);
                        }
                    }
            }
        }
    }
    __syncthreads();

    // ---------------- Phase B: WMMA matmul  h = w1 x folded ---------------
    // D(16x16) = A(16x4: w1 block) x B(4x16: folded block), K = 64 in 16 steps
    const int wv = tid >> 5;
    const int lane = tid & 31;
    const int l  = lane & 15;
    const int hf = lane >> 4;                // lane half (0/1)
    const int o0 = (wv & 3) * 16;            // output-channel block

    #pragma unroll
    for (int t = 0; t < 8; ++t) {
        const int pb   = (wv >> 2) + 2 * t;  // pixel block 0..15
        const int pix0 = pb * 16;
        v8f acc = {0.f, 0.f, 0.f, 0.f, 0.f, 0.f, 0.f, 0.f};
        #pragma unroll
        for (int c0 = 0; c0 < CH; c0 += 4) {
            // A frag: VGPR0 = K 2*hf, VGPR1 = K 2*hf+1 ; M = l
            v2f a, bb;
            a.x = w1[(o0 + l) * CH + c0 + 2 * hf];
            a.y = w1[(o0 + l) * CH + c0 + 2 * hf + 1];
            // B frag: VGPR0 = row 2*hf, VGPR1 = row 2*hf+1 ; N = l
            bb.x = s_fold[(c0 + 2 * hf) * 256 + pix0 + l];
            bb.y = s_fold[(c0 + 2 * hf + 1) * 256 + pix0 + l];
            acc = __builtin_amdgcn_wmma_f32_16x16x4_f32(
                false, a, false, bb, (short)0, acc, false, false);
        }
        // D layout: vgpr v, lanes0-15 -> (M=v, N=l); lanes16-31 -> (M=v+8)
        #pragma unroll
        for (int v = 0; v < 8; ++v) {
            const int o = o0 + v + 8 * hf;
            float hv = acc[v] + b1[o];
            hv = (hv >= 0.0f) ? hv : 0.1f * hv;          // LeakyReLU(0.1)
            s_h[(pix0 + l) * 68 + o] = hv;
        }
    }
    __syncthreads();

    // coalesced store: one thread per pixel, 64 contiguous floats
    {
        const int p  = tid;
        const int hh = th0 + (p >> 4);
        const int ww = tw0 + (p & 15);
        float* hp = hws + ((size_t)(b * HT + hh) * WT + ww) * CH;
        #pragma unroll
        for (int i = 0; i < 16; ++i) {
            v4f v = *(const v4f*)(&s_h[p * 68 + i * 4]);
            *(v4f*)(hp + i * 4) = v;
        }
    }
}

// ---------------------------------------------------------------------------
// Kernel 2: 3x3 conv (64 -> 3) + b2 + border-sampled residual of inp -> out
// grid = (144, B), block = 256 (one thread per output pixel)
// ---------------------------------------------------------------------------
__global__ __launch_bounds__(256) void k_conv_out(
    const float* __restrict__ hws,     // (B,192,192,64)
    const float* __restrict__ coord,   // (B,192,192,2)
    const float* __restrict__ inp,     // (B,3,96,96)
    const float* __restrict__ w2,      // (3,64,3,3)
    const float* __restrict__ b2,      // (3)
    float* __restrict__ out)           // (B,3,192,192)
{
    __shared__ float s_w2[3 * CH * 9];
    __shared__ float s_b2[3];
    const int tid = threadIdx.x;
    for (int i = tid; i < 3 * CH * 9; i += 256) s_w2[i] = w2[i];
    if (tid < 3) s_b2[tid] = b2[tid];
    __syncthreads();

    const int b   = blockIdx.y;
    const int pix = blockIdx.x * 256 + tid;
    const int h   = pix / WT;
    const int w   = pix % WT;

    float a0 = s_b2[0], a1 = s_b2[1], a2 = s_b2[2];
    #pragma unroll
    for (int dy = 0; dy < 3; ++dy) {
        const int hy = h + dy - 1;
        if (hy < 0 || hy >= HT) continue;
        #pragma unroll
        for (int dx = 0; dx < 3; ++dx) {
            const int wx = w + dx - 1;
            if (wx < 0 || wx >= WT) continue;
            const float* hp = hws + ((size_t)(b * HT + hy) * WT + wx) * CH;
            const int widx = dy * 3 + dx;
            #pragma unroll
            for (int c0 = 0; c0 < CH; c0 += 4) {
                v4f v = *(const v4f*)(hp + c0);
                #pragma unroll
                for (int j = 0; j < 4; ++j) {
                    const int c = c0 + j;
                    a0 += s_w2[(0 * CH + c) * 9 + widx] * v[j];
                    a1 += s_w2[(1 * CH + c) * 9 + widx] * v[j];
                    a2 += s_w2[(2 * CH + c) * 9 + widx] * v[j];
                }
            }
        }
    }

    // residual: bilinear border sample of inp at reversed coords
    const float sy = coord[((size_t)(b * HT + h) * WT + w) * 2 + 0];
    const float sx = coord[((size_t)(b * HT + h) * WT + w) * 2 + 1];
    float ix = ((sx + 1.0f) * (float)WIN - 1.0f) * 0.5f;
    float iy = ((sy + 1.0f) * (float)HIN - 1.0f) * 0.5f;
    ix = fminf(fmaxf(ix, 0.0f), (float)(WIN - 1));
    iy = fminf(fmaxf(iy, 0.0f), (float)(HIN - 1));
    const float fx0 = floorf(ix), fy0 = floorf(iy);
    const float wx1 = ix - fx0,  wy1 = iy - fy0;
    const float wx0 = 1.0f - wx1, wy0 = 1.0f - wy1;
    const int x0 = (int)fx0;
    const int y0 = (int)fy0;
    const int x1 = (x0 + 1 < WIN) ? x0 + 1 : WIN - 1;
    const int y1 = (y0 + 1 < HIN) ? y0 + 1 : HIN - 1;
    const float* ib = inp + (size_t)b * 3 * HIN * WIN;
    float r[3];
    #pragma unroll
    for (int ch = 0; ch < 3; ++ch) {
        const float* ic = ib + (size_t)ch * HIN * WIN;
        r[ch] = wy0 * (wx0 * ic[y0 * WIN + x0] + wx1 * ic[y0 * WIN + x1]) +
                wy1 * (wx0 * ic[y1 * WIN + x0] + wx1 * ic[y1 * WIN + x1]);
    }

    const size_t ob = (size_t)b * 3 * HT * WT + (size_t)pix;
    out[ob]               = a0 + r[0];
    out[ob + HT * WT]     = a1 + r[1];
    out[ob + 2 * HT * WT] = a2 + r[2];
}

// ---------------------------------------------------------------------------
// Launch
// ---------------------------------------------------------------------------
extern "C" void kernel_launch(void* const* d_in, const int* in_sizes, int n_in,
                              void* d_out, int out_size, void* d_ws, size_t ws_size,
                              hipStream_t stream) {
    const float* feat  = (const float*)d_in[0];   // (4,64,96,96)
    const float* kern  = (const float*)d_in[1];   // (4,576,36864)
    const float* coord = (const float*)d_in[2];   // (4,192,192,2)
    const float* inp   = (const float*)d_in[3];   // (4,3,96,96)
    const float* w1    = (const float*)d_in[4];   // (64,64)
    const float* b1    = (const float*)d_in[5];   // (64)
    const float* w2    = (const float*)d_in[6];   // (3,64,3,3)
    const float* b2    = (const float*)d_in[7];   // (3)
    float* out = (float*)d_out;

    float* featT = (float*)d_ws;                               // 2,359,296 f
    float* hws   = featT + (size_t)BATCH * HIN * WIN * CH;     // 9,437,184 f

    k_transpose_feat<<<dim3(WIN / 32, HIN, BATCH), 256, 0, stream>>>(feat, featT);
    k_fused_main<<<dim3(WT / TILE, HT / TILE, BATCH), 256, 0, stream>>>(
        featT, kern, coord, w1, b1, hws);
    k_conv_out<<<dim3((HT * WT) / 256, BATCH), 256, 0, stream>>>(
        hws, coord, inp, w2, b2, out);
}